// Cross_Attention_17798344474862
// MI455X (gfx1250) — compile-verified
//
#include <hip/hip_runtime.h>
#include <hip/hip_bf16.h>

#define BB   8
#define NN   16384      // H*W
#define CC   64
#define OC   128        // 2*C
#define LN_EPS 1e-5f

typedef __attribute__((ext_vector_type(2))) float v2f;
typedef __attribute__((ext_vector_type(8))) float v8f;

// ---------------- Kernel 1: column-softmax partial stats (per slab of 1024 tokens) ----
// grid (B, 16), block 256 = (c=tid&63, j=tid>>6)
__global__ __launch_bounds__(256) void k_colstats_partial(const float* __restrict__ x2,
                                                          float* __restrict__ pm,
                                                          float* __restrict__ ps) {
    int b = blockIdx.x, slab = blockIdx.y;
    int c = threadIdx.x & 63, j = threadIdx.x >> 6;
    const float* base = x2 + ((size_t)b * NN + (size_t)slab * 1024) * CC + c;
    float m = -INFINITY;
    #pragma unroll 4
    for (int t = 0; t < 256; ++t) {
        float v = base[(size_t)(j + 4 * t) * CC];
        m = fmaxf(m, v);
    }
    float s = 0.f;
    #pragma unroll 4
    for (int t = 0; t < 256; ++t) {
        float v = base[(size_t)(j + 4 * t) * CC];
        s += __expf(v - m);
    }
    __shared__ float lm[4][64], ls[4][64];
    lm[j][c] = m; ls[j][c] = s;
    __syncthreads();
    if (j == 0) {
        float M = lm[0][c];
        for (int q = 1; q < 4; ++q) M = fmaxf(M, lm[q][c]);
        float S = 0.f;
        for (int q = 0; q < 4; ++q) S += ls[q][c] * __expf(lm[q][c] - M);
        pm[(b * 16 + slab) * 64 + c] = M;
        ps[(b * 16 + slab) * 64 + c] = S;
    }
}

// ---------------- Kernel 2: combine slab stats -> colmax, 1/colsum -------------------
__global__ void k_colstats_combine(const float* __restrict__ pm, const float* __restrict__ ps,
                                   float* __restrict__ cmax, float* __restrict__ crcp) {
    int i = blockIdx.x * blockDim.x + threadIdx.x;   // 0..511 = b*64+c
    if (i >= BB * CC) return;
    int b = i >> 6, c = i & 63;
    float M = -INFINITY;
    for (int s = 0; s < 16; ++s) M = fmaxf(M, pm[(b * 16 + s) * 64 + c]);
    float S = 0.f;
    for (int s = 0; s < 16; ++s) S += ps[(b * 16 + s) * 64 + c] * __expf(pm[(b * 16 + s) * 64 + c] - M);
    cmax[i] = M;
    crcp[i] = 1.f / S;
}

// ---------------- Kernel 3: row-softmax stats (one wave per token row) ----------------
// grid B*N/8, block 256 (8 waves)
__global__ __launch_bounds__(256) void k_rowstats(const float* __restrict__ x2,
                                                  float* __restrict__ rmax,
                                                  float* __restrict__ rrcp) {
    int wave = threadIdx.x >> 5, lane = threadIdx.x & 31;
    size_t row = (size_t)blockIdx.x * 8 + wave;      // < B*N
    const float* p = x2 + row * CC;
    float v0 = p[lane], v1 = p[lane + 32];
    float m = fmaxf(v0, v1);
    for (int off = 16; off; off >>= 1) m = fmaxf(m, __shfl_xor(m, off, 32));
    float s = __expf(v0 - m) + __expf(v1 - m);
    for (int off = 16; off; off >>= 1) s += __shfl_xor(s, off, 32);
    if (lane == 0) { rmax[row] = m; rrcp[row] = 1.f / s; }
}

// ---------------- Kernel 4: sim partials = q^T k over a K-slab of 512 tokens ---------
// grid (B, 32), block 256 (8 waves, 2 tiles each). WMMA f32 16x16x4, K = tokens.
__global__ __launch_bounds__(256) void k_sim(const float* __restrict__ x2,
                                             const float* __restrict__ rmax,
                                             const float* __restrict__ rrcp,
                                             const float* __restrict__ cmax,
                                             const float* __restrict__ crcp,
                                             float* __restrict__ simpart) {
    int b = blockIdx.x, slab = blockIdx.y;
    int wave = threadIdx.x >> 5, lane = threadIdx.x & 31;
    int lm = lane & 15, lh = lane >> 4;
    const int nbase = slab * 512;
    const float* X  = x2   + (size_t)b * NN * CC;
    const float* RM = rmax + (size_t)b * NN;
    const float* RR = rrcp + (size_t)b * NN;

    for (int ti = 0; ti < 2; ++ti) {
        int tile = wave + ti * 8;                 // 0..15
        int c0 = (tile >> 2) * 16, d0 = (tile & 3) * 16;
        float cmd = cmax[b * 64 + d0 + lm];
        float crd = crcp[b * 64 + d0 + lm];
        v8f acc = {};
        for (int kk = 0; kk < 512; kk += 4) {
            int n = nbase + kk + 2 * lh;
            float rm0 = RM[n],     rr0 = RR[n];
            float rm1 = RM[n + 1], rr1 = RR[n + 1];
            float xa0 = X[(size_t)n * CC + c0 + lm];
            float xa1 = X[(size_t)(n + 1) * CC + c0 + lm];
            float xb0 = X[(size_t)n * CC + d0 + lm];
            float xb1 = X[(size_t)(n + 1) * CC + d0 + lm];
            v2f a, bv;
            a.x  = __expf(xa0 - rm0) * rr0;       // query = softmax over C
            a.y  = __expf(xa1 - rm1) * rr1;
            bv.x = __expf(xb0 - cmd) * crd;       // key   = softmax over N
            bv.y = __expf(xb1 - cmd) * crd;
            acc = __builtin_amdgcn_wmma_f32_16x16x4_f32(false, a, false, bv,
                                                        (short)0, acc, false, false);
        }
        #pragma unroll
        for (int r = 0; r < 8; ++r) {
            int cc = c0 + r + 8 * lh;             // D layout: M = r + 8*half
            simpart[(((size_t)slab * BB + b) * 64 + cc) * 64 + d0 + lm] = acc[r];
        }
    }
}

// ---------------- Kernel 4r: reduce 32 slab partials ---------------------------------
__global__ void k_simreduce(const float* __restrict__ simpart, float* __restrict__ sim) {
    int i = blockIdx.x * blockDim.x + threadIdx.x;   // < B*64*64
    if (i >= BB * 64 * 64) return;
    float s = 0.f;
    for (int sl = 0; sl < 32; ++sl) s += simpart[(size_t)sl * BB * 4096 + i];
    sim[i] = s;
}

// ---------------- Kernel 5: M2 = conv_w @ sim  [B,128,64] -----------------------------
__global__ __launch_bounds__(128) void k_m2(const float* __restrict__ conv_w,
                                            const float* __restrict__ sim,
                                            float* __restrict__ M2) {
    int b = blockIdx.x, o = threadIdx.x;             // o in 0..127
    const float* W = conv_w + o * 64;
    const float* S = sim + (size_t)b * 4096;
    float acc[64];
    #pragma unroll
    for (int c = 0; c < 64; ++c) acc[c] = 0.f;
    for (int d = 0; d < 64; ++d) {
        float wd = W[d];
        const float* Sr = S + d * 64;
        #pragma unroll
        for (int c = 0; c < 64; ++c) acc[c] += wd * Sr[c];
    }
    float* out = M2 + ((size_t)b * 128 + o) * 64;
    #pragma unroll
    for (int c = 0; c < 64; ++c) out[c] = acc[c];
}

// ---------------- Kernel 6: out = LN( M2 @ x1^T + conv_b ) ----------------------------
// grid (N/128, B), block 256 (8 waves; wave -> 16-token slab x all 128 out channels)
__global__ __launch_bounds__(256) void k_out(const float* __restrict__ x1,
                                             const float* __restrict__ M2,
                                             const float* __restrict__ conv_b,
                                             const float* __restrict__ gamma,
                                             const float* __restrict__ beta,
                                             float* __restrict__ out) {
    int b = blockIdx.y;
    int wave = threadIdx.x >> 5, lane = threadIdx.x & 31;
    int lm = lane & 15, lh = lane >> 4;
    int n0 = blockIdx.x * 128 + wave * 16;

    __shared__ float lM2[128 * 68];                  // +4 pad: conflict-free f2 reads
    const float* M2b = M2 + (size_t)b * 8192;
    for (int i = threadIdx.x; i < 8192; i += 256) {
        int o = i >> 6, c = i & 63;
        lM2[o * 68 + c] = M2b[i];
    }
    __syncthreads();

    float g[8], bt[8], cb[8];
    #pragma unroll
    for (int t = 0; t < 8; ++t) {
        int o = t * 16 + lm;
        g[t] = gamma[o]; bt[t] = beta[o]; cb[t] = conv_b[o];
    }

    v8f acc[8];
    #pragma unroll
    for (int t = 0; t < 8; ++t) acc[t] = {};

    const float* A = x1 + ((size_t)b * NN + n0) * CC;
    for (int kk = 0; kk < 64; kk += 4) {
        v2f a = *(const v2f*)(A + (size_t)lm * CC + kk + 2 * lh);
        #pragma unroll
        for (int t = 0; t < 8; ++t) {
            v2f bv = *(const v2f*)(&lM2[(t * 16 + lm) * 68 + kk + 2 * lh]);
            acc[t] = __builtin_amdgcn_wmma_f32_16x16x4_f32(false, a, false, bv,
                                                           (short)0, acc[t], false, false);
        }
    }

    // conv bias
    #pragma unroll
    for (int t = 0; t < 8; ++t)
        #pragma unroll
        for (int r = 0; r < 8; ++r) acc[t][r] += cb[t];

    // LayerNorm over 128 channels: token m = r + 8*lh lives in 16 lanes of this half
    float mu[8], rstd[8];
    #pragma unroll
    for (int r = 0; r < 8; ++r) {
        float s = 0.f;
        #pragma unroll
        for (int t = 0; t < 8; ++t) s += acc[t][r];
        s += __shfl_xor(s, 1, 32); s += __shfl_xor(s, 2, 32);
        s += __shfl_xor(s, 4, 32); s += __shfl_xor(s, 8, 32);
        mu[r] = s * (1.f / 128.f);
    }
    #pragma unroll
    for (int r = 0; r < 8; ++r) {
        float s = 0.f;
        #pragma unroll
        for (int t = 0; t < 8; ++t) { float d = acc[t][r] - mu[r]; s += d * d; }
        s += __shfl_xor(s, 1, 32); s += __shfl_xor(s, 2, 32);
        s += __shfl_xor(s, 4, 32); s += __shfl_xor(s, 8, 32);
        rstd[r] = rsqrtf(s * (1.f / 128.f) + LN_EPS);
    }

    float* O = out + ((size_t)b * NN + n0) * OC;
    #pragma unroll
    for (int t = 0; t < 8; ++t)
        #pragma unroll
        for (int r = 0; r < 8; ++r) {
            int m = r + 8 * lh;
            O[(size_t)m * OC + t * 16 + lm] = (acc[t][r] - mu[r]) * rstd[r] * g[t] + bt[t];
        }
}

// ---------------- Host launcher -------------------------------------------------------
extern "C" void kernel_launch(void* const* d_in, const int* in_sizes, int n_in,
                              void* d_out, int out_size, void* d_ws, size_t ws_size,
                              hipStream_t stream) {
    (void)in_sizes; (void)n_in; (void)out_size; (void)ws_size;
    const float* x1     = (const float*)d_in[0];
    const float* x2     = (const float*)d_in[1];
    const float* conv_w = (const float*)d_in[2];
    const float* conv_b = (const float*)d_in[3];
    const float* lng    = (const float*)d_in[4];
    const float* lnb    = (const float*)d_in[5];
    float* out = (float*)d_out;

    float* ws = (float*)d_ws;
    float* pm      = ws;                 // B*16*64      = 8192
    float* ps      = ws + 8192;          //              = 8192
    float* cmax    = ws + 16384;         // B*64         = 512
    float* crcp    = ws + 16896;         //              = 512
    float* rmax    = ws + 17408;         // B*N          = 131072
    float* rrcp    = ws + 148480;        //              = 131072
    float* sim     = ws + 279552;        // B*64*64      = 32768
    float* M2      = ws + 312320;        // B*128*64     = 65536
    float* simpart = ws + 377856;        // 32*B*64*64   = 1048576  (total ~5.7 MB)

    k_colstats_partial<<<dim3(BB, 16), 256, 0, stream>>>(x2, pm, ps);
    k_colstats_combine<<<2, 256, 0, stream>>>(pm, ps, cmax, crcp);
    k_rowstats<<<(BB * NN) / 8, 256, 0, stream>>>(x2, rmax, rrcp);
    k_sim<<<dim3(BB, 32), 256, 0, stream>>>(x2, rmax, rrcp, cmax, crcp, simpart);
    k_simreduce<<<(BB * 4096 + 255) / 256, 256, 0, stream>>>(simpart, sim);
    k_m2<<<BB, 128, 0, stream>>>(conv_w, sim, M2);
    k_out<<<dim3(NN / 128, BB), 256, 0, stream>>>(x1, M2, conv_b, lng, lnb, out);
}